// DS1_87582973100610
// MI455X (gfx1250) — compile-verified
//
#include <hip/hip_runtime.h>

typedef __attribute__((ext_vector_type(2))) float v2f;
typedef __attribute__((ext_vector_type(4))) float v4f;
typedef __attribute__((ext_vector_type(8))) float v8f;

#define DDIM 64           // feature dim
#define UDIM 64           // number of prototypes
#define ROWS_PER_WG 128   // 8 waves * 16 rows
#define XPITCH 68         // padded LDS row pitch (floats) -> conflict-free b64 frag reads
#define WPITCH 68

// out[n,u] = ||x_n||^2 - 2 x_n . w_u + ||w_u||^2
__global__ __launch_bounds__(256, 2)
void sqdist_wmma_f32_kernel(const float* __restrict__ x,
                            const float* __restrict__ w,
                            float* __restrict__ out,
                            int nrows) {
  __shared__ float lx[ROWS_PER_WG * XPITCH];   // 34816 B
  __shared__ float lw[UDIM * WPITCH];          // 17408 B
  __shared__ float lx2[ROWS_PER_WG];
  __shared__ float lw2[UDIM];

  const int tid  = threadIdx.x;
  const int row0 = blockIdx.x * ROWS_PER_WG;

  // ---- Stage X tile: 128x64 f32, coalesced float4 loads (8 per thread) ----
#pragma unroll
  for (int i = 0; i < 8; ++i) {
    int g  = i * 256 + tid;        // float4 index within the tile
    int r  = g >> 4;               // row within tile (16 float4 per row)
    int c4 = g & 15;
    int rg = row0 + r;
    if (rg >= nrows) rg = nrows - 1;           // clamp (tail safety)
    v4f v = *(const v4f*)(x + (size_t)rg * DDIM + c4 * 4);
    *(v4f*)&lx[r * XPITCH + c4 * 4] = v;       // 272B row pitch: 16B aligned
  }
  // ---- Stage W: 64x64 f32 (4 float4 per thread) ----
#pragma unroll
  for (int i = 0; i < 4; ++i) {
    int g  = i * 256 + tid;
    int r  = g >> 4;
    int c4 = g & 15;
    v4f v = *(const v4f*)(w + (size_t)r * DDIM + c4 * 4);
    *(v4f*)&lw[r * WPITCH + c4 * 4] = v;
  }
  __syncthreads();

  // ---- Row norms: ||x_r||^2 (threads 0..127), ||w_u||^2 (threads 128..191) ----
  if (tid < ROWS_PER_WG) {
    const float* p = &lx[tid * XPITCH];
    float s = 0.f;
#pragma unroll
    for (int d = 0; d < DDIM; ++d) s = fmaf(p[d], p[d], s);
    lx2[tid] = s;
  } else if (tid < ROWS_PER_WG + UDIM) {
    const float* p = &lw[(tid - ROWS_PER_WG) * WPITCH];
    float s = 0.f;
#pragma unroll
    for (int d = 0; d < DDIM; ++d) s = fmaf(p[d], p[d], s);
    lw2[tid - ROWS_PER_WG] = s;
  }
  __syncthreads();

  // ---- WMMA main loop: each wave computes a 16x64 output tile ----
  const int wave = tid >> 5;
  const int lane = tid & 31;
  const int m    = lane & 15;     // A: row / B: column / C: column index
  const int hi   = lane >> 4;     // lane half selects K pair (A/B) or M+8 (C)
  const int wrow = wave * 16;     // wave's first row within the tile

  // A frag: lane holds {X[m][kk+2*hi], X[m][kk+2*hi+1]}  (ISA 16x4 f32 layout)
  const float* ax  = &lx[(wrow + m) * XPITCH + 2 * hi];
  // B frag: lane holds {W[u0+m][kk+2*hi], W[u0+m][kk+2*hi+1]} (B = W^T, 4x16 layout)
  const float* bw0 = &lw[( 0 + m) * WPITCH + 2 * hi];
  const float* bw1 = &lw[(16 + m) * WPITCH + 2 * hi];
  const float* bw2 = &lw[(32 + m) * WPITCH + 2 * hi];
  const float* bw3 = &lw[(48 + m) * WPITCH + 2 * hi];

  v8f acc0 = {}, acc1 = {}, acc2 = {}, acc3 = {};

#pragma unroll
  for (int kk = 0; kk < DDIM; kk += 4) {
    v2f a  = *(const v2f*)(ax  + kk);   // ds_load_b64, conflict-free (68-float pitch)
    v2f q0 = *(const v2f*)(bw0 + kk);
    v2f q1 = *(const v2f*)(bw1 + kk);
    v2f q2 = *(const v2f*)(bw2 + kk);
    v2f q3 = *(const v2f*)(bw3 + kk);
    acc0 = __builtin_amdgcn_wmma_f32_16x16x4_f32(false, a, false, q0, (short)0, acc0, false, false);
    acc1 = __builtin_amdgcn_wmma_f32_16x16x4_f32(false, a, false, q1, (short)0, acc1, false, false);
    acc2 = __builtin_amdgcn_wmma_f32_16x16x4_f32(false, a, false, q2, (short)0, acc2, false, false);
    acc3 = __builtin_amdgcn_wmma_f32_16x16x4_f32(false, a, false, q3, (short)0, acc3, false, false);
  }

  // ---- Epilogue: out = x2 + w2 - 2*dot ----
  // C/D layout: VGPR v -> row (wrow + v + 8*hi), column (u_tile*16 + m)
#define EPILOGUE(ACC, UT)                                                     \
  do {                                                                        \
    float w2v = lw2[(UT) * 16 + m];                                           \
    _Pragma("unroll")                                                         \
    for (int v = 0; v < 8; ++v) {                                             \
      int r = wrow + v + 8 * hi;                                              \
      float res = fmaf(-2.0f, (ACC)[v], lx2[r] + w2v);                        \
      int rg = row0 + r;                                                      \
      if (rg < nrows)                                                         \
        out[(size_t)rg * UDIM + (UT) * 16 + m] = res;                         \
    }                                                                         \
  } while (0)

  EPILOGUE(acc0, 0);
  EPILOGUE(acc1, 1);
  EPILOGUE(acc2, 2);
  EPILOGUE(acc3, 3);
#undef EPILOGUE
}

extern "C" void kernel_launch(void* const* d_in, const int* in_sizes, int n_in,
                              void* d_out, int out_size, void* d_ws, size_t ws_size,
                              hipStream_t stream) {
  (void)n_in; (void)out_size; (void)d_ws; (void)ws_size;
  const float* x = (const float*)d_in[0];   // [B,H,W,D] f32, flat N x 64
  const float* w = (const float*)d_in[1];   // [U,D] f32, 64 x 64
  float* out = (float*)d_out;               // [N,U] f32

  const int nrows  = in_sizes[0] / DDIM;    // 524288 for the reference shapes
  const int blocks = (nrows + ROWS_PER_WG - 1) / ROWS_PER_WG;
  sqdist_wmma_f32_kernel<<<blocks, 256, 0, stream>>>(x, w, out, nrows);
}